// YOLOLoss_81423989998151
// MI455X (gfx1250) — compile-verified
//
#include <hip/hip_runtime.h>
#include <hip/hip_bf16.h>
#include <math.h>

#define NCLS 7
#define NANC 3

typedef float v2f_t __attribute__((ext_vector_type(2)));
typedef float v8f_t __attribute__((ext_vector_type(8)));

// Exact fp32 wave(32) sum via chained V_WMMA_F32_16X16X4_F32.
// A = {p, 0}: lanes 0-15 supply A[m][0]=p_m, lanes 16-31 supply A[m][2]=p_{m+16}.
// B = all ones -> D[m][n] = p_m + p_{m+16}.
// C/D layout: VGPR v, lanes 0-15 hold M=v, lanes 16-31 hold M=v+8.
// Summing c[0..7] per lane gives half-wave sums; second WMMA adds both halves,
// leaving the full 32-lane sum in every lane's c2[0].
__device__ __forceinline__ float wave_sum_wmma(float p) {
    v2f_t a;    a.x = p;    a.y = 0.0f;
    v2f_t ones; ones.x = 1.0f; ones.y = 1.0f;
    v8f_t c = {0.f, 0.f, 0.f, 0.f, 0.f, 0.f, 0.f, 0.f};
    c = __builtin_amdgcn_wmma_f32_16x16x4_f32(false, a, false, ones, (short)0, c,
                                              false, false);
    float s = ((c[0] + c[1]) + (c[2] + c[3])) + ((c[4] + c[5]) + (c[6] + c[7]));
    v2f_t a2; a2.x = s; a2.y = 0.0f;
    v8f_t c2 = {0.f, 0.f, 0.f, 0.f, 0.f, 0.f, 0.f, 0.f};
    c2 = __builtin_amdgcn_wmma_f32_16x16x4_f32(false, a2, false, ones, (short)0, c2,
                                               false, false);
    return c2[0];
}

// bce-with-logits element, z == 0 case: max(x,0) + log1p(exp(-|x|))
__device__ __forceinline__ float bce0(float x) {
    return fmaxf(x, 0.0f) + log1pf(expf(-fabsf(x)));
}

// Workspace layout:
//   [0, 128)   : 16 doubles (accumulators): [0..2]=iou sums, [3..5]=cls sums,
//                [6..8]=obj bce sums (incl. -x dedup adjustments)
//   [128, ...) : bit flags, 1 bit per (b,a,gy,gx) cell, per-scale word bases
//                s0: 0, s1: 38400, s2: 48000  (total 50400 words)
#define ACC_DOUBLES 16
#define FLAG_WORDS  50400
#define ZERO_WORDS  (32 + FLAG_WORDS)   // 32 u32 == 16 doubles

__global__ void yolo_zero(unsigned int* __restrict__ ws, int nwords) {
    int i = blockIdx.x * blockDim.x + threadIdx.x;
    if (i < nwords) ws[i] = 0u;
}

__global__ void yolo_targets(const float* __restrict__ p0,
                             const float* __restrict__ p1,
                             const float* __restrict__ p2,
                             const float* __restrict__ tg,
                             const float* __restrict__ anc,
                             double* __restrict__ acc,
                             unsigned int* __restrict__ flags,
                             int nt) {
    int gid = blockIdx.x * blockDim.x + threadIdx.x;
    if (gid >= nt * 3) return;
    int s = gid / nt;
    int t = gid - s * nt;

    const float* pred = (s == 0) ? p0 : ((s == 1) ? p1 : p2);
    int W        = (s == 0) ? 80 : ((s == 1) ? 40 : 20);
    int HW       = W * W;
    int flagBase = (s == 0) ? 0 : ((s == 1) ? 38400 : 48000);

    const float* row = tg + (size_t)t * 6;
    int   b   = (int)row[0];
    int   cls = (int)row[1];
    float gx  = row[2] * (float)W;
    float gy  = row[3] * (float)W;   // H == W
    float gw  = row[4] * (float)W;
    float gh  = row[5] * (float)W;

    // anchor argmax (first-max tie-break, matching jnp.argmax)
    float best = -1e30f;
    int   j = 0;
#pragma unroll
    for (int a = 0; a < NANC; ++a) {
        float aw = anc[s * 6 + a * 2 + 0];
        float ah = anc[s * 6 + a * 2 + 1];
        float r1 = gw / aw, r2 = gh / ah;
        float v = fminf(fminf(r1, 1.0f / r1), fminf(r2, 1.0f / r2));
        if (v > best) { best = v; j = a; }
    }

    int gxi = (int)gx; gxi = gxi < 0 ? 0 : (gxi > W - 1 ? W - 1 : gxi);
    int gyi = (int)gy; gyi = gyi < 0 ? 0 : (gyi > W - 1 ? W - 1 : gyi);

    // gather pg = pred[b, j*12+ch, gy, gx]
    const float* base = pred + (size_t)(b * 36 + j * 12) * HW + (size_t)gyi * W + gxi;
    float pg[12];
#pragma unroll
    for (int ch = 0; ch < 12; ++ch) pg[ch] = base[(size_t)ch * HW];

    // 1 - IoU(pg[:4], box_t)
    float tx = gx - (float)gxi, ty = gy - (float)gyi;
    float b1x1 = pg[0] - pg[2] * 0.5f, b1y1 = pg[1] - pg[3] * 0.5f;
    float b1x2 = pg[0] + pg[2] * 0.5f, b1y2 = pg[1] + pg[3] * 0.5f;
    float b2x1 = tx - gw * 0.5f, b2y1 = ty - gh * 0.5f;
    float b2x2 = tx + gw * 0.5f, b2y2 = ty + gh * 0.5f;
    float iw = fmaxf(fminf(b1x2, b2x2) - fmaxf(b1x1, b2x1), 0.0f);
    float ih = fmaxf(fminf(b1y2, b2y2) - fmaxf(b1y1, b2y1), 0.0f);
    float inter = iw * ih;
    float a1 = (b1x2 - b1x1) * (b1y2 - b1y1);
    float a2 = (b2x2 - b2x1) * (b2y2 - b2y1);
    float iou = inter / (a1 + a2 - inter + 1e-16f);
    atomicAdd(&acc[0 + s], (double)(1.0f - iou));

    // class BCE over 7 logits vs one-hot
    float csum = 0.0f;
#pragma unroll
    for (int k = 0; k < NCLS; ++k) {
        float x = pg[5 + k];
        float z = (k == cls) ? 1.0f : 0.0f;
        csum += fmaxf(x, 0.0f) - x * z + log1pf(expf(-fabsf(x)));
    }
    atomicAdd(&acc[3 + s], (double)csum);

    // dedup obj-cell adjustment: bce(x,1)-bce(x,0) == -x, once per unique cell
    int cell = ((b * 3 + j) * W + gyi) * W + gxi;
    unsigned int* wptr = flags + flagBase + (cell >> 5);
    unsigned int  bit  = 1u << (cell & 31);
    unsigned int  old  = atomicOr(wptr, bit);
    if (!(old & bit)) atomicAdd(&acc[6 + s], (double)(-pg[4]));
}

__global__ void yolo_obj(const float* __restrict__ p0,
                         const float* __restrict__ p1,
                         const float* __restrict__ p2,
                         double* __restrict__ acc,
                         int B) {
    int s = blockIdx.y;
    const float* pred = (s == 0) ? p0 : ((s == 1) ? p1 : p2);
    int HW  = (s == 0) ? 6400 : ((s == 1) ? 1600 : 400);
    int HW4 = HW >> 2;
    int n4  = B * 3 * HW4;

    int tid    = blockIdx.x * blockDim.x + threadIdx.x;
    int stride = gridDim.x * blockDim.x;

    float partial = 0.0f;
    for (int i = tid; i < n4; i += stride) {
        int plane = i / HW4;             // b*3 + a
        int r     = i - plane * HW4;
        int b     = plane / 3;
        int a     = plane - b * 3;
        const float4* ptr =
            (const float4*)(pred + (size_t)(b * 36 + a * 12 + 4) * HW) + r;
        float4 x = *ptr;
        partial += (bce0(x.x) + bce0(x.y)) + (bce0(x.z) + bce0(x.w));
    }
    // EXEC is all-ones here (divergence reconverged); full waves guaranteed
    float ws = wave_sum_wmma(partial);
    if ((threadIdx.x & 31) == 0) atomicAdd(&acc[6 + s], (double)ws);
}

__global__ void yolo_final(const double* __restrict__ acc,
                           float* __restrict__ out, int B, int nt) {
    double lbox = (acc[0] + acc[1] + acc[2]) / (double)nt;
    double lcls = (acc[3] + acc[4] + acc[5]) / (double)(nt * NCLS);
    double n0 = (double)B * 3.0 * 6400.0;
    double n1 = (double)B * 3.0 * 1600.0;
    double n2 = (double)B * 3.0 * 400.0;
    double lobj = acc[6] / n0 * 4.0 + acc[7] / n1 * 1.0 + acc[8] / n2 * 0.4;
    out[0] = (float)(0.05 * lbox + 1.0 * lobj + 0.5 * lcls);
}

extern "C" void kernel_launch(void* const* d_in, const int* in_sizes, int n_in,
                              void* d_out, int out_size, void* d_ws, size_t ws_size,
                              hipStream_t stream) {
    const float* p0  = (const float*)d_in[0];
    const float* p1  = (const float*)d_in[1];
    const float* p2  = (const float*)d_in[2];
    const float* tg  = (const float*)d_in[3];
    const float* anc = (const float*)d_in[4];

    double*       acc   = (double*)d_ws;
    unsigned int* flags = (unsigned int*)((char*)d_ws + 128);

    int nt = in_sizes[3] / 6;                 // 1000
    int B  = in_sizes[0] / (36 * 6400);       // 64

    yolo_zero<<<(ZERO_WORDS + 255) / 256, 256, 0, stream>>>(
        (unsigned int*)d_ws, ZERO_WORDS);

    int ttot = nt * 3;
    yolo_targets<<<(ttot + 127) / 128, 128, 0, stream>>>(
        p0, p1, p2, tg, anc, acc, flags, nt);

    yolo_obj<<<dim3(256, 3), 256, 0, stream>>>(p0, p1, p2, acc, B);

    yolo_final<<<1, 1, 0, stream>>>(acc, (float*)d_out, B, nt);
}